// DCConv3dKernelUnitPolynomials_71382356459673
// MI455X (gfx1250) — compile-verified
//
#include <hip/hip_runtime.h>
#include <hip/hip_bf16.h>
#include <math.h>

typedef __attribute__((ext_vector_type(2))) float v2f;
typedef __attribute__((ext_vector_type(8))) float v8f;

// out[p] = sum_i c_i * R_{n_i,l_i}(r) * Y_{l_i,m_i}(theta,phi) over the 30-term
// hydrogenic basis (N=4,K=3,M=3). The 32-long (zero-padded) contraction is done
// with V_WMMA_F32_16X16X4_F32: A = coefficients broadcast across all 16 rows,
// B = per-point psi chunk (column n = point slot). One wave32 covers 16 points;
// the lane half supplies the K={0,1} vs K={2,3} slices (K = vgpr + 2*half).
// Inputs are range-bounded (r<10, theta<pi, phi<2pi) so hardware transcendental
// paths (__sinf/__cosf/__expf -> v_sin/v_cos/v_exp) are accurate and cheap.
__global__ __launch_bounds__(256) void hydro_basis_wmma_kernel(
    const float* __restrict__ pos,   // (npts, 3) = r, theta, phi
    const float* __restrict__ coef,  // (30,)
    float* __restrict__ out,         // (npts,)
    int npts)
{
    const int lane = threadIdx.x & 31;
    const int wave = threadIdx.x >> 5;
    const bool hb  = (lane & 16) != 0;   // lane half -> K pair selector
    const int m    = lane & 15;          // point slot within wave

    const int p  = blockIdx.x * 128 + wave * 16 + m;
    const int pc = (p < npts) ? p : (npts - 1);   // clamp: keep EXEC all-ones for WMMA

    const float* P = pos + 3 * pc;
    const float r  = P[0];
    const float th = P[1];
    const float ph = P[2];

    // ---- angular transcendentals + multiple angles --------------------------
    const float st = __sinf(th);
    const float ct = __cosf(th);
    const float sp = __sinf(ph);
    const float cp = __cosf(ph);
    const float s2p = 2.0f * sp * cp;
    const float c2p = fmaf(cp, cp, -(sp * sp));
    const float s3p = fmaf(s2p, cp,  c2p * sp);
    const float c3p = fmaf(c2p, cp, -(s2p * sp));
    const float ct2 = ct * ct;
    const float st2 = st * st;
    const float st3 = st2 * st;

    // ---- radial functions R_{nl}(r), rho_n = 2r/n ---------------------------
    const float e1 = __expf(-r);
    const float e2 = __expf(-0.5f * r);
    const float e3 = __expf(-r * (1.0f / 3.0f));
    const float e4 = __expf(-0.25f * r);
    const float rho3 = r * (2.0f / 3.0f);
    const float rho4 = r * 0.5f;

    const float R10 = 2.0f * e1;
    const float R20 = 0.35355339f * (2.0f - r) * e2;
    const float R21 = 0.20412415f * r * e2;
    const float R30 = 0.12830006f * fmaf(rho3, fmaf(rho3, 0.5f, -3.0f), 3.0f) * e3;
    const float R31 = 0.045360921f * rho3 * (4.0f - rho3) * e3;
    const float R32 = 0.020286021f * rho3 * rho3 * e3;
    const float q40 = fmaf(rho4, fmaf(rho4, fmaf(rho4, -1.0f / 6.0f, 2.0f), -6.0f), 4.0f);
    const float R40 = 0.0625f * q40 * e4;
    const float R41 = 0.016137431f * rho4 * fmaf(rho4, fmaf(rho4, 0.5f, -5.0f), 10.0f) * e4;
    const float R42 = 0.0046584751f * rho4 * rho4 * (6.0f - rho4) * e4;
    const float R43 = 0.0017607380f * rho4 * rho4 * rho4 * e4;

    // ---- real spherical harmonics (Condon-Shortley, reference normalization)
    const float Y00  = 0.28209479f;
    const float g1   = 0.48860251f * st;
    const float Y1m1 = -g1 * sp;
    const float Y10  = 0.48860251f * ct;
    const float Y11  = -g1 * cp;
    const float g2a  = 0.54627422f * st2;
    const float Y2m2 = g2a * s2p;
    const float Y22  = g2a * c2p;
    const float g2b  = 1.09254843f * st * ct;
    const float Y2m1 = -g2b * sp;
    const float Y21  = -g2b * cp;
    const float Y20  = fmaf(ct2, 0.94617471f, -0.31539157f);
    const float g3a  = 0.59004359f * st3;
    const float Y3m3 = -g3a * s3p;
    const float Y33  = -g3a * c3p;
    const float g3b  = 1.44530572f * st2 * ct;
    const float Y3m2 = g3b * s2p;
    const float Y32  = g3b * c2p;
    const float g3c  = 0.45704580f * st * fmaf(ct2, 5.0f, -1.0f);
    const float Y3m1 = -g3c * sp;
    const float Y31  = -g3c * cp;
    const float Y30  = ct * fmaf(ct2, 1.86588166f, -1.11952899f);

    // ---- psi values (reference basis order), individually named -------------
    const float p0  = R10 * Y00;
    const float p1  = R20 * Y00;
    const float p2  = R21 * Y1m1;
    const float p3  = R21 * Y10;
    const float p4  = R21 * Y11;
    const float p5  = R30 * Y00;
    const float p6  = R31 * Y1m1;
    const float p7  = R31 * Y10;
    const float p8  = R31 * Y11;
    const float p9  = R32 * Y2m2;
    const float p10 = R32 * Y2m1;
    const float p11 = R32 * Y20;
    const float p12 = R32 * Y21;
    const float p13 = R32 * Y22;
    const float p14 = R40 * Y00;
    const float p15 = R41 * Y1m1;
    const float p16 = R41 * Y10;
    const float p17 = R41 * Y11;
    const float p18 = R42 * Y2m2;
    const float p19 = R42 * Y2m1;
    const float p20 = R42 * Y20;
    const float p21 = R42 * Y21;
    const float p22 = R42 * Y22;
    const float p23 = R43 * Y3m3;
    const float p24 = R43 * Y3m2;
    const float p25 = R43 * Y3m1;
    const float p26 = R43 * Y30;
    const float p27 = R43 * Y31;
    const float p28 = R43 * Y32;
    const float p29 = R43 * Y33;

    // Coefficients: uniform pointer + constant indices -> scalar loads (SGPRs).
    const float c0  = coef[0],  c1  = coef[1],  c2  = coef[2],  c3  = coef[3];
    const float c4  = coef[4],  c5  = coef[5],  c6  = coef[6],  c7  = coef[7];
    const float c8  = coef[8],  c9  = coef[9],  c10 = coef[10], c11 = coef[11];
    const float c12 = coef[12], c13 = coef[13], c14 = coef[14], c15 = coef[15];
    const float c16 = coef[16], c17 = coef[17], c18 = coef[18], c19 = coef[19];
    const float c20 = coef[20], c21 = coef[21], c22 = coef[22], c23 = coef[23];
    const float c24 = coef[24], c25 = coef[25], c26 = coef[26], c27 = coef[27];
    const float c28 = coef[28], c29 = coef[29];

    // ---- D = A x B + C over 8 chunks of K=4 ---------------------------------
    // Each operand element: one v_cndmask between two named scalars.
    v8f acc = {0.0f, 0.0f, 0.0f, 0.0f, 0.0f, 0.0f, 0.0f, 0.0f};

#define HB_CHUNK(cA, cB, cC, cD, pA, pB, pC, pD)                               \
    {                                                                          \
        v2f a, b;                                                              \
        a.x = hb ? (cC) : (cA);                                                \
        a.y = hb ? (cD) : (cB);                                                \
        b.x = hb ? (pC) : (pA);                                                \
        b.y = hb ? (pD) : (pB);                                                \
        acc = __builtin_amdgcn_wmma_f32_16x16x4_f32(                           \
            false, a, false, b, (short)0, acc, false, false);                  \
    }

    HB_CHUNK(c0,  c1,  c2,  c3,  p0,  p1,  p2,  p3)
    HB_CHUNK(c4,  c5,  c6,  c7,  p4,  p5,  p6,  p7)
    HB_CHUNK(c8,  c9,  c10, c11, p8,  p9,  p10, p11)
    HB_CHUNK(c12, c13, c14, c15, p12, p13, p14, p15)
    HB_CHUNK(c16, c17, c18, c19, p16, p17, p18, p19)
    HB_CHUNK(c20, c21, c22, c23, p20, p21, p22, p23)
    HB_CHUNK(c24, c25, c26, c27, p24, p25, p26, p27)
    HB_CHUNK(c28, c29, 0.0f, 0.0f, p28, p29, 0.0f, 0.0f)
#undef HB_CHUNK

    if (!hb && p < npts) {
        out[p] = acc[0];   // D row M=0, column N=m  (lanes 0..15, VGPR0)
    }
}

extern "C" void kernel_launch(void* const* d_in, const int* in_sizes, int n_in,
                              void* d_out, int out_size, void* d_ws, size_t ws_size,
                              hipStream_t stream) {
    (void)n_in; (void)d_ws; (void)ws_size;
    const float* pos  = (const float*)d_in[0];   // (4096, 512, 3) float32
    const float* coef = (const float*)d_in[1];   // (30,) float32
    float* out        = (float*)d_out;           // (4096, 512) float32

    const int npts = out_size;                   // 2,097,152
    const int blocks = (npts + 127) / 128;       // 128 points per 256-thread block
    hydro_basis_wmma_kernel<<<blocks, 256, 0, stream>>>(pos, coef, out, npts);
}